// ScenePrototypeBank_19473381720435
// MI455X (gfx1250) — compile-verified
//
#include <hip/hip_runtime.h>
#include <math.h>

typedef __attribute__((ext_vector_type(2))) float v2f;
typedef __attribute__((ext_vector_type(4))) float v4f;
typedef __attribute__((ext_vector_type(8))) float v8f;
typedef __attribute__((ext_vector_type(4))) unsigned int u32x4;
typedef __attribute__((ext_vector_type(4))) int i32x4;
typedef __attribute__((ext_vector_type(8))) int i32x8;

#define TAU    0.2f
#define EPS_N  1e-8f
#define LOG2E  1.44269504088896340736f
#define LDS_PITCH 68   // 64 + 4 floats: strided float2 gather hits 64 distinct banks

#define HAVE_TDM __has_builtin(__builtin_amdgcn_tensor_load_to_lds)

// ---------------------------------------------------------------------------
// Main kernel: one wave per 16-row tile, 4 waves per block (64 rows/block).
//  - features: per-wave TDM (tensor_load_to_lds) DMA of a 16x64 f32 tile into
//    LDS with native row padding to pitch 68 (bank-conflict-free A gather)
//  - prototypes: one coalesced global->LDS stage per block (pitch 68)
//  - logits tile [16x64]: 4 x 16x16 WMMA accumulators over K=64
//  - softmax: in-register + xor-shuffle reduction over the 16-lane col group
// ---------------------------------------------------------------------------
__global__ __launch_bounds__(128) void proto_assign_wmma(
    const float* __restrict__ F,     // [N,64] features
    const float* __restrict__ P,     // [64,64] prototypes (rows L2-normed)
    float* __restrict__ outL,        // [N,64] logits
    float* __restrict__ outS)        // [N,64] softmax probs
{
    __shared__ float Psh[64 * LDS_PITCH];                 // prototypes, padded
    __shared__ float Fsh[4 * 16 * LDS_PITCH];             // per-wave feature tiles

    const int tid  = threadIdx.x;
    const int lane = tid & 31;
    const int wave = tid >> 5;
    const int m    = lane & 15;   // A: tile row   | B/D: tile col
    const int h    = lane >> 4;   // half selector
    const int row0 = (blockIdx.x * 4 + wave) * 16;

#if HAVE_TDM
    // ---- issue per-wave TDM: DMA 16 rows x 64 f32 -> LDS (pitch 68) --------
    {
        const unsigned long long ga =
            (unsigned long long)(uintptr_t)(F + (size_t)row0 * 64);
        const unsigned int la =
            (unsigned int)(uintptr_t)&Fsh[wave * 16 * LDS_PITCH];

        u32x4 g0;
        g0.x = 1u;                                     // count=1 (valid D#)
        g0.y = la;                                     // lds_addr (bytes)
        g0.z = (unsigned int)ga;                       // global_addr[31:0]
        g0.w = (unsigned int)((ga >> 32) & 0x1FFFFFFu) // global_addr[56:32]
             | (2u << 30);                             // type = 2 ("image")

        i32x8 g1;
        g1[0] = (2 << 16)      // data_size = 4 bytes
              | (1 << 20)      // pad_enable
              | (5 << 22)      // pad_interval: every 64 DWORDs (2^(5+1))
              | (3 << 25);     // pad_amount: 4 DWORDs -> row pitch 68
        g1[1] = 64 << 16;      // tensor_dim0 = 64 elems (bits 79:48, low half)
        g1[2] = 16 << 16;      // tensor_dim1 = 16 rows  (bits 111:80, low half)
        g1[3] = 64 << 16;      // tile_dim0   = 64 elems (bits 127:112)
        g1[4] = 16;            // tile_dim1   = 16 rows  (bits 143:128)
        g1[5] = 64;            // tensor_dim0_stride = 64 elems (bits 207:160)
        g1[6] = 0;
        g1[7] = 0;

        i32x4 gz = {};         // 2D tensor: groups 2/3 unused
#if __clang_major__ >= 23
        i32x8 gz8 = {};
        __builtin_amdgcn_tensor_load_to_lds(g0, g1, gz, gz, gz8, 0);
#else
        __builtin_amdgcn_tensor_load_to_lds(g0, g1, gz, gz, 0);
#endif
    }
#endif

    // ---- stage prototypes to LDS (padded pitch), coalesced b128 loads ------
    for (int i = tid; i < 64 * 16; i += 128) {
        const int row = i >> 4;
        const int c4  = (i & 15) << 2;
        v4f v = *(const v4f*)(P + row * 64 + c4);
        *(v4f*)&Psh[row * LDS_PITCH + c4] = v;
    }
    __syncthreads();
#if HAVE_TDM
    __builtin_amdgcn_s_wait_tensorcnt(0);   // own wave's feature tile landed
#endif

    // ---- gather 16 feature rows in WMMA-A (16x4 f32) layout ----------------
    // step k covers K=4k..4k+3; lane (m,h) holds {feat[m][4k+2h], feat[m][4k+2h+1]}
    v2f a[16];
#if HAVE_TDM
    const float* frow = &Fsh[(wave * 16 + m) * LDS_PITCH + 2 * h];
#else
    const float* frow = F + (size_t)(row0 + m) * 64 + 2 * h;
#endif
#pragma unroll
    for (int k = 0; k < 16; ++k)
        a[k] = *(const v2f*)(frow + 4 * k);

    // ---- row L2 norm: each lane holds one half (32 elems) of its row -------
    float ss = 0.f;
#pragma unroll
    for (int k = 0; k < 16; ++k)
        ss += a[k].x * a[k].x + a[k].y * a[k].y;
    ss += __shfl_xor(ss, 16, 32);                 // combine the two halves
    const float nrm   = sqrtf(ss);
    const float scale = 1.0f / (fmaxf(nrm, EPS_N) * TAU);  // fold 1/tau in
#pragma unroll
    for (int k = 0; k < 16; ++k) { a[k].x *= scale; a[k].y *= scale; }

    // ---- logits tile: 4 col-tiles x 16 K-steps of V_WMMA_F32_16X16X4_F32 ---
    v8f acc[4];
#pragma unroll
    for (int c = 0; c < 4; ++c) {
        v8f d = {};
        const float* bbase = &Psh[(16 * c + m) * LDS_PITCH + 2 * h];
#pragma unroll
        for (int k = 0; k < 16; ++k) {
            v2f b = *(const v2f*)(bbase + 4 * k);
            d = __builtin_amdgcn_wmma_f32_16x16x4_f32(
                    /*neg_a=*/false, a[k], /*neg_b=*/false, b,
                    /*c_mod=*/(short)0, d, /*reuse_a=*/false, /*reuse_b=*/false);
        }
        acc[c] = d;
    }

    // ---- softmax over 64 cols per row; lane holds rows r+8h, col 16c+m -----
#pragma unroll
    for (int r = 0; r < 8; ++r) {
        float x0 = acc[0][r], x1 = acc[1][r], x2 = acc[2][r], x3 = acc[3][r];
        float mx = fmaxf(fmaxf(x0, x1), fmaxf(x2, x3));
        mx = fmaxf(mx, __shfl_xor(mx, 1, 32));
        mx = fmaxf(mx, __shfl_xor(mx, 2, 32));
        mx = fmaxf(mx, __shfl_xor(mx, 4, 32));
        mx = fmaxf(mx, __shfl_xor(mx, 8, 32));
        float e0 = exp2f((x0 - mx) * LOG2E);
        float e1 = exp2f((x1 - mx) * LOG2E);
        float e2 = exp2f((x2 - mx) * LOG2E);
        float e3 = exp2f((x3 - mx) * LOG2E);
        float s = e0 + e1 + e2 + e3;
        s += __shfl_xor(s, 1, 32);
        s += __shfl_xor(s, 2, 32);
        s += __shfl_xor(s, 4, 32);
        s += __shfl_xor(s, 8, 32);
        const float inv = 1.0f / s;

        const size_t base = (size_t)(row0 + r + 8 * h) * 64 + m;
        outL[base +  0] = x0;
        outL[base + 16] = x1;
        outL[base + 32] = x2;
        outL[base + 48] = x3;
        outS[base +  0] = e0 * inv;
        outS[base + 16] = e1 * inv;
        outS[base + 32] = e2 * inv;
        outS[base + 48] = e3 * inv;
    }
}

// ---------------------------------------------------------------------------
// Scalar tail for N % 64 != 0 (never launched for the reference N=1e6).
// ---------------------------------------------------------------------------
__global__ void proto_assign_tail(
    const float* __restrict__ F, const float* __restrict__ P,
    float* __restrict__ outL, float* __restrict__ outS, int start, int nTotal)
{
    const int row = start + blockIdx.x * blockDim.x + threadIdx.x;
    if (row >= nTotal) return;

    float x[64];
    float ss = 0.f;
#pragma clang loop unroll(disable)
    for (int d = 0; d < 64; ++d) { x[d] = F[(size_t)row * 64 + d]; ss += x[d] * x[d]; }
    const float scale = 1.0f / (fmaxf(sqrtf(ss), EPS_N) * TAU);

    float l[64];
    float mx = -INFINITY;
#pragma clang loop unroll(disable)
    for (int j = 0; j < 64; ++j) {
        float acc = 0.f;
#pragma clang loop unroll(disable)
        for (int d = 0; d < 64; ++d) acc += x[d] * P[j * 64 + d];
        l[j] = acc * scale;
        mx = fmaxf(mx, l[j]);
    }
    float s = 0.f;
#pragma clang loop unroll(disable)
    for (int j = 0; j < 64; ++j) s += exp2f((l[j] - mx) * LOG2E);
    const float inv = 1.0f / s;
#pragma clang loop unroll(disable)
    for (int j = 0; j < 64; ++j) {
        outL[(size_t)row * 64 + j] = l[j];
        outS[(size_t)row * 64 + j] = exp2f((l[j] - mx) * LOG2E) * inv;
    }
}

extern "C" void kernel_launch(void* const* d_in, const int* in_sizes, int n_in,
                              void* d_out, int out_size, void* d_ws, size_t ws_size,
                              hipStream_t stream) {
    const float* F = (const float*)d_in[0];   // [N,64] fp32
    const float* P = (const float*)d_in[1];   // [64,64] fp32
    float* out  = (float*)d_out;
    const int N = in_sizes[0] / 64;
    float* outL = out;
    float* outS = out + (size_t)N * 64;

    const int nBlocks = N / 64;               // 64 rows per block, exact for N=1e6
    if (nBlocks > 0)
        proto_assign_wmma<<<nBlocks, 128, 0, stream>>>(F, P, outL, outS);

    const int rem = N - nBlocks * 64;
    if (rem > 0)
        proto_assign_tail<<<(rem + 63) / 64, 64, 0, stream>>>(F, P, outL, outS,
                                                              nBlocks * 64, N);
}